// TripletLoss_18202071400492
// MI455X (gfx1250) — compile-verified
//
#include <hip/hip_runtime.h>
#include <hip/hip_bf16.h>
#include <math.h>

typedef float v2f __attribute__((ext_vector_type(2)));
typedef float v8f __attribute__((ext_vector_type(8)));

#define NROWS 8192
#define DIM   128
#define MARGIN 0.3f
#define EPSF   1e-12f

// ---------------- Kernel 1: L2-normalize rows, record sq = ||x||^2 (post-norm) ----
__global__ __launch_bounds__(256) void tl_normalize(const float* __restrict__ X,
                                                    float* __restrict__ Xn,
                                                    float* __restrict__ sq) {
  const int wave = threadIdx.x >> 5;
  const int lane = threadIdx.x & 31;
  const int row  = blockIdx.x * 8 + wave;           // 8 waves per block, 1 row per wave
  float4 v = ((const float4*)(X + row * DIM))[lane]; // 32 lanes * 4 = 128
  float s = v.x * v.x + v.y * v.y + v.z * v.z + v.w * v.w;
  #pragma unroll
  for (int m = 16; m >= 1; m >>= 1) s += __shfl_xor(s, m, 32);
  const float norm = sqrtf(s);
  const float inv  = 1.0f / (norm + EPSF);
  float4 o;
  o.x = v.x * inv; o.y = v.y * inv; o.z = v.z * inv; o.w = v.w * inv;
  ((float4*)(Xn + row * DIM))[lane] = o;
  if (lane == 0) sq[row] = s * inv * inv;           // == sum((x*inv)^2)
}

// ---------------- Kernel 2: fused X*X^T (WMMA f32) + hard mining --------------
// 4 waves/block; each wave owns a 16-row block; block loops over 64-col panels in LDS.
#define TM 64          // rows per block
#define PN 64          // panel columns
#define LSTR 132       // padded LDS row stride (dwords): conflict-free & 16B aligned

__global__ __launch_bounds__(128) void tl_pairwise_hardmine(
    const float* __restrict__ Xn, const float* __restrict__ sq,
    const int* __restrict__ targets,
    float* __restrict__ ap_out, float* __restrict__ an_out) {
  __shared__ float Xp[PN * LSTR];
  __shared__ float sqp[PN];
  __shared__ int   labp[PN];

  const int tid  = threadIdx.x;
  const int wave = tid >> 5;
  const int lane = tid & 31;
  const int half = lane >> 4;       // 0: lanes 0-15, 1: lanes 16-31
  const int l16  = lane & 15;
  const int koff = half * 2;        // K-pair select per WMMA f32 16x16x4 layout

  const int i0 = blockIdx.x * TM + wave * 16;   // this wave's 16-row tile

  // Preload A fragments for all 32 K-steps (resident in VGPRs).
  v2f a[32];
  {
    const int arow = i0 + l16;
    #pragma unroll
    for (int ks = 0; ks < 32; ++ks)
      a[ks] = *(const v2f*)(Xn + arow * DIM + ks * 4 + koff);
  }

  // Per-lane metadata for the 8 C rows this lane holds (m = half*8 + r).
  float sqi[8]; int labi[8];
  #pragma unroll
  for (int r = 0; r < 8; ++r) {
    const int gr = i0 + half * 8 + r;
    sqi[r]  = sq[gr];
    labi[r] = targets[gr];
  }

  float ap[8], an[8];
  #pragma unroll
  for (int r = 0; r < 8; ++r) { ap[r] = -__builtin_inff(); an[r] = __builtin_inff(); }

  for (int j0 = 0; j0 < NROWS; j0 += PN) {
    __syncthreads();
    // Stage 64x128 fp32 panel into LDS (padded rows), plus sq/labels.
    #pragma unroll
    for (int it = 0; it < 16; ++it) {
      const int q    = it * 128 + tid;     // 0..2047 float4 chunks
      const int prow = q >> 5;
      const int pcol = (q & 31) * 4;
      float4 v = *(const float4*)(Xn + (j0 + prow) * DIM + pcol);
      *(float4*)&Xp[prow * LSTR + pcol] = v;
    }
    if (tid < PN) { sqp[tid] = sq[j0 + tid]; labp[tid] = targets[j0 + tid]; }
    __syncthreads();

    #pragma unroll
    for (int jt = 0; jt < 4; ++jt) {
      v8f acc = {};
      const float* bbase = &Xp[(jt * 16 + l16) * LSTR + koff];
      #pragma unroll
      for (int ks = 0; ks < 32; ++ks) {
        v2f b = *(const v2f*)(bbase + ks * 4);
        acc = __builtin_amdgcn_wmma_f32_16x16x4_f32(
            /*neg_a=*/false, a[ks], /*neg_b=*/false, b,
            /*c_mod=*/(short)0, acc, /*reuse_a=*/false, /*reuse_b=*/false);
      }
      // Fused distance + hard mining on the 16x16 tile (no materialization).
      const int   jn   = jt * 16 + l16;
      const float sqj  = sqp[jn];
      const int   labj = labp[jn];
      #pragma unroll
      for (int r = 0; r < 8; ++r) {
        const float d2 = sqi[r] + sqj - 2.0f * acc[r];
        const float d  = sqrtf(fmaxf(d2, EPSF));
        const bool pos = (labi[r] == labj);
        ap[r] = pos ? fmaxf(ap[r], d) : ap[r];
        an[r] = pos ? an[r] : fminf(an[r], d);
      }
    }
  }

  // Reduce across the 16 lanes sharing each C row (xor stays within half).
  #pragma unroll
  for (int m = 1; m <= 8; m <<= 1) {
    #pragma unroll
    for (int r = 0; r < 8; ++r) {
      ap[r] = fmaxf(ap[r], __shfl_xor(ap[r], m, 32));
      an[r] = fminf(an[r], __shfl_xor(an[r], m, 32));
    }
  }
  if (l16 == 0) {
    #pragma unroll
    for (int r = 0; r < 8; ++r) {
      const int gr = i0 + half * 8 + r;
      ap_out[gr] = ap[r];
      an_out[gr] = an[r];
    }
  }
}

// ---------------- Kernel 3: loss = mean(relu(ap - an + margin)) ----------------
__global__ __launch_bounds__(256) void tl_reduce(const float* __restrict__ ap,
                                                 const float* __restrict__ an,
                                                 float* __restrict__ out) {
  __shared__ float wsum[8];
  float s = 0.0f;
  for (int i = threadIdx.x; i < NROWS; i += 256)
    s += fmaxf(ap[i] - an[i] + MARGIN, 0.0f);
  #pragma unroll
  for (int m = 16; m >= 1; m >>= 1) s += __shfl_xor(s, m, 32);
  if ((threadIdx.x & 31) == 0) wsum[threadIdx.x >> 5] = s;
  __syncthreads();
  if (threadIdx.x == 0) {
    float t = 0.0f;
    #pragma unroll
    for (int w = 0; w < 8; ++w) t += wsum[w];
    out[0] = t / (float)NROWS;
  }
}

extern "C" void kernel_launch(void* const* d_in, const int* in_sizes, int n_in,
                              void* d_out, int out_size, void* d_ws, size_t ws_size,
                              hipStream_t stream) {
  const float* X       = (const float*)d_in[0];   // [8192,128] fp32
  const int*   targets = (const int*)d_in[1];     // [8192] int32
  float* ws = (float*)d_ws;
  float* Xn = ws;                                 // 8192*128 floats
  float* sq = Xn + (size_t)NROWS * DIM;           // 8192
  float* ap = sq + NROWS;                         // 8192
  float* an = ap + NROWS;                         // 8192

  tl_normalize<<<NROWS / 8, 256, 0, stream>>>(X, Xn, sq);
  tl_pairwise_hardmine<<<NROWS / TM, 128, 0, stream>>>(Xn, sq, targets, ap, an);
  tl_reduce<<<1, 256, 0, stream>>>(ap, an, (float*)d_out);
}